// Closs_58978490909000
// MI455X (gfx1250) — compile-verified
//
#include <hip/hip_runtime.h>

namespace {
constexpr int kB    = 8192;   // rows
constexpr int kN    = 2048;   // cols
constexpr int kH    = 1024;   // half
constexpr int kTPB  = 256;    // threads per block (8 wave32)
constexpr int kWaves = kTPB / 32;
}

// One block per row. Computes
//   res = [sum(f[H:]) - sum(f[:H])]
//       + sum_{i=0..H-1} [ LSE(f[i:N-i]) + LSE(-f[i:N-i]) ]
// via per-row max + suffix-scan identity, scaled by 1/B.
__global__ __launch_bounds__(kTPB) void closs_rows(const float* __restrict__ f,
                                                   float* __restrict__ partial,
                                                   float* __restrict__ out_atomic) {
  __shared__ float  row[kN];          // 8 KB staged row
  __shared__ float4 wred[kWaves];     // pass-1 wave partials (sum, max, min)
  __shared__ float2 wtot[kWaves];     // scan wave totals (p,q)
  __shared__ float  wacc[kWaves];     // final acc wave partials

  const int t    = threadIdx.x;
  const int lane = t & 31;
  const int wave = t >> 5;
  const float* rp = f + (size_t)blockIdx.x * kN;

  // ---- CDNA5 async global->LDS staging of the row (ASYNCcnt path) ----
  // Each thread issues two B128 (16B) async copies: 256 thr * 32 B = 8 KB.
  {
    unsigned lds0 = (unsigned)(size_t)&row[0] + (unsigned)(t * 16);
    unsigned lds1 = lds0 + (unsigned)(kTPB * 16);
    unsigned long long g0 = (unsigned long long)(size_t)(rp + t * 4);
    unsigned long long g1 = g0 + (unsigned long long)(kTPB * 16);
    asm volatile("global_load_async_to_lds_b128 %0, %1, off" :: "v"(lds0), "v"(g0) : "memory");
    asm volatile("global_load_async_to_lds_b128 %0, %1, off" :: "v"(lds1), "v"(g1) : "memory");
    asm volatile("s_wait_asynccnt 0" ::: "memory");
  }
  __syncthreads();  // barrier #1: row visible to all waves

  // ---- Pass 1: signed sum / max / min, wave32 shuffle butterfly ----
  float lsum = 0.f, mx = -3.402823466e38f, mn = 3.402823466e38f;
#pragma unroll
  for (int k = 0; k < 8; ++k) {
    float v = row[t * 8 + k];
    lsum += v;
    mx = fmaxf(mx, v);
    mn = fminf(mn, v);
  }
  // threads t<128 cover j<1024 (negative sign), t>=128 cover j>=1024 (positive)
  float ls = (t >= kTPB / 2) ? lsum : -lsum;
#pragma unroll
  for (int d = 16; d > 0; d >>= 1) {
    ls += __shfl_xor(ls, d, 32);
    mx = fmaxf(mx, __shfl_xor(mx, d, 32));
    mn = fminf(mn, __shfl_xor(mn, d, 32));
  }
  if (lane == 0) wred[wave] = make_float4(ls, mx, mn, 0.f);
  __syncthreads();  // barrier #2
  float lterm = 0.f, MpAll = -3.402823466e38f, mnAll = 3.402823466e38f;
#pragma unroll
  for (int w = 0; w < kWaves; ++w) {
    float4 a = wred[w];
    lterm += a.x;
    MpAll = fmaxf(MpAll, a.y);
    mnAll = fminf(mnAll, a.z);
  }
  const float Mp = MpAll;   // max f
  const float Mn = -mnAll;  // max -f

  // ---- Pass 2: pair sums p[i], q[i]; local suffix sums in registers ----
  // thread t owns i = 4t .. 4t+3 (wave w owns i in [128w, 128w+128))
  float tp[4], tq[4];
  {
    float p[4], q[4];
#pragma unroll
    for (int k = 0; k < 4; ++k) {
      int i = t * 4 + k;
      float a = row[i];
      float b = row[kN - 1 - i];
      p[k] = __expf(a - Mp) + __expf(b - Mp);
      q[k] = __expf(-a - Mn) + __expf(-b - Mn);
    }
    tp[3] = p[3];          tq[3] = q[3];
    tp[2] = p[2] + tp[3];  tq[2] = q[2] + tq[3];
    tp[1] = p[1] + tp[2];  tq[1] = q[1] + tq[2];
    tp[0] = p[0] + tp[1];  tq[0] = q[0] + tq[1];
  }
  const float sp = tp[0];  // per-thread totals
  const float sq = tq[0];

  // ---- Intra-wave inclusive suffix scan (shfl_down Hillis-Steele) ----
  float ip = sp, iq = sq;
#pragma unroll
  for (int d = 1; d < 32; d <<= 1) {
    float op = __shfl_down(ip, d, 32);
    float oq = __shfl_down(iq, d, 32);
    if (lane + d < 32) { ip += op; iq += oq; }
  }
  // lane 0 now holds the whole-wave total
  if (lane == 0) wtot[wave] = make_float2(ip, iq);
  __syncthreads();  // barrier #3
  // cross-wave exclusive suffix offset (fixed ascending order: deterministic)
  float offp = 0.f, offq = 0.f;
#pragma unroll
  for (int w = 0; w < kWaves; ++w) {
    if (w > wave) { offp += wtot[w].x; offq += wtot[w].y; }
  }
  // thread-level exclusive suffix offset
  offp += ip - sp;
  offq += iq - sq;

  // ---- log of window sums; 2 logs per window index ----
  float acc = 0.f;
#pragma unroll
  for (int k = 0; k < 4; ++k) {
    acc += __logf(offp + tp[k]);
    acc += __logf(offq + tq[k]);
  }

  // ---- final reduce: shuffle butterfly + 8-entry fold ----
#pragma unroll
  for (int d = 16; d > 0; d >>= 1) acc += __shfl_xor(acc, d, 32);
  if (lane == 0) wacc[wave] = acc;
  __syncthreads();  // barrier #4
  if (t == 0) {
    float atot = 0.f;
#pragma unroll
    for (int w = 0; w < kWaves; ++w) atot += wacc[w];
    float res = lterm + (float)kH * (Mp + Mn) + atot;
    res *= (1.0f / (float)kB);
    if (partial) partial[blockIdx.x] = res;
    else         atomicAdd(out_atomic, res);
  }
}

// Deterministic fixed-order reduction of the 8192 per-row partials.
__global__ __launch_bounds__(1024) void closs_final(const float* __restrict__ partial,
                                                    float* __restrict__ out) {
  __shared__ float buf[1024];
  const int t = threadIdx.x;
  float s = 0.f;
#pragma unroll
  for (int k = 0; k < kB / 1024; ++k) s += partial[t + k * 1024];
  buf[t] = s;
  __syncthreads();
  for (int d = 512; d > 0; d >>= 1) {
    if (t < d) buf[t] += buf[t + d];
    __syncthreads();
  }
  if (t == 0) out[0] = buf[0];
}

__global__ void closs_zero(float* out) { out[0] = 0.f; }

extern "C" void kernel_launch(void* const* d_in, const int* in_sizes, int n_in,
                              void* d_out, int out_size, void* d_ws, size_t ws_size,
                              hipStream_t stream) {
  const float* f = (const float*)d_in[0];  // (8192, 2048) f32
  float* out = (float*)d_out;              // scalar f32

  if (ws_size >= (size_t)kB * sizeof(float)) {
    float* partial = (float*)d_ws;
    closs_rows<<<kB, kTPB, 0, stream>>>(f, partial, nullptr);
    closs_final<<<1, 1024, 0, stream>>>(partial, out);
  } else {
    // fallback: atomic accumulation (still correct; order may vary)
    closs_zero<<<1, 1, 0, stream>>>(out);
    closs_rows<<<kB, kTPB, 0, stream>>>(f, nullptr, out);
  }
}